// DirectedGCNConv_34256659153348
// MI455X (gfx1250) — compile-verified
//
#include <hip/hip_runtime.h>

typedef __attribute__((ext_vector_type(2))) float v2f;
typedef __attribute__((ext_vector_type(8))) float v8f;

#define D_FEAT 64

// ---------- small utility kernels ----------

__global__ void k_fill1(float* __restrict__ p, int n) {
  int i = blockIdx.x * blockDim.x + threadIdx.x;
  if (i < n) p[i] = 1.0f;
}

__global__ void k_zero4(float4* __restrict__ p, int n4) {
  int i = blockIdx.x * blockDim.x + threadIdx.x;
  if (i < n4) p[i] = make_float4(0.f, 0.f, 0.f, 0.f);
}

__global__ void k_rsqrt(float* __restrict__ p, int n) {
  int i = blockIdx.x * blockDim.x + threadIdx.x;
  if (i < n) p[i] = rsqrtf(p[i]);
}

// One pass over the edges builds both in-degree (keyed on dst) and
// out-degree (keyed on src) histograms. Buffers pre-filled with 1.0 (self-loop).
__global__ void k_count_deg(const long long* __restrict__ esrc,
                            const long long* __restrict__ edst,
                            float* __restrict__ deg_f,   // direction 1: col = dst
                            float* __restrict__ deg_b,   // direction 2: col = src
                            int E) {
  int e = blockIdx.x * blockDim.x + threadIdx.x;
  if (e < E) {
    int s = (int)esrc[e];
    int d = (int)edst[e];
    atomicAdd(&deg_f[d], 1.0f);
    atomicAdd(&deg_b[s], 1.0f);
  }
}

// ---------- WMMA fp32 GEMM: H[N,64] = X[N,64] @ W[64,64] ----------
// One wave computes a 16x16 tile of H; 4 waves per block cover the 64 columns.
// Layouts per CDNA5 ISA 7.12.2 (32-bit A 16x4, B/C/D row-striped across lanes).
__global__ __launch_bounds__(128)
void k_gemm_wmma_f32(const float* __restrict__ X,
                     const float* __restrict__ W,
                     float* __restrict__ H) {
  const int lane = threadIdx.x & 31;
  const int wave = threadIdx.x >> 5;
  const int m    = lane & 15;   // row within tile (A), col within tile (B/C/D)
  const int kh   = lane >> 4;   // K-half selector (0/1) within a K=4 step
  const int row0 = blockIdx.x * 16;
  const int n0   = wave * 16;

  v8f c = {};
  const float* xrow = X + (size_t)(row0 + m) * D_FEAT;

#pragma unroll
  for (int k0 = 0; k0 < D_FEAT; k0 += 4) {
    // A 16x4 fp32: lane holds M=m, VGPR v -> K = k0 + v + 2*kh  (contiguous pair)
    v2f a = *(const v2f*)(xrow + k0 + 2 * kh);
    // B 4x16 fp32: VGPR v -> K = k0 + v + 2*kh, N = n0 + m
    v2f b;
    b.x = W[(size_t)(k0 + 2 * kh + 0) * D_FEAT + n0 + m];
    b.y = W[(size_t)(k0 + 2 * kh + 1) * D_FEAT + n0 + m];
    c = __builtin_amdgcn_wmma_f32_16x16x4_f32(false, a, false, b,
                                              (short)0, c, false, false);
  }

  // C/D 16x16 fp32: VGPR v -> M = v + 8*kh, N = n0 + m
  float* hrow = H + (size_t)(row0 + 8 * kh) * D_FEAT + n0 + m;
#pragma unroll
  for (int v = 0; v < 8; ++v)
    hrow[(size_t)v * D_FEAT] = c[v];
}

// ---------- edge scatter: acc[col] += h[row] * dinv[row]*dinv[col] ----------
// 16 lanes per edge, each lane handles 4 contiguous features (float4 gather,
// 4 global fp32 atomics). 16 lanes * 16B = one fully coalesced 256B h-row read.
__global__ void k_scatter(const long long* __restrict__ erow,
                          const long long* __restrict__ ecol,
                          const float* __restrict__ dinv,
                          const float* __restrict__ H,
                          float* __restrict__ acc, int E) {
  long long t = (long long)blockIdx.x * blockDim.x + threadIdx.x;
  int e = (int)(t >> 4);
  int q = (int)(t & 15);
  if (e < E) {
    int r = (int)erow[e];
    int c = (int)ecol[e];
    float norm = dinv[r] * dinv[c];
    const float4 hv = *(const float4*)(H + (size_t)r * D_FEAT + q * 4);
    float* ap = acc + (size_t)c * D_FEAT + q * 4;
    atomicAdd(ap + 0, hv.x * norm);
    atomicAdd(ap + 1, hv.y * norm);
    atomicAdd(ap + 2, hv.z * norm);
    atomicAdd(ap + 3, hv.w * norm);
  }
}

// Self-loop edge (i,i): acc[i] += h[i] * dinv[i]^2. Runs as its own kernel
// (stream-ordered after the scatter), so plain adds suffice — no atomics.
__global__ void k_selfloop(const float* __restrict__ dinv,
                           const float* __restrict__ H,
                           float* __restrict__ acc, int n) {
  int t = blockIdx.x * blockDim.x + threadIdx.x;
  int i = t >> 4;
  int q = t & 15;
  if (i < n) {
    float w = dinv[i] * dinv[i];
    const float4 hv = *(const float4*)(H + (size_t)i * D_FEAT + q * 4);
    float4* ap = (float4*)(acc + (size_t)i * D_FEAT + q * 4);
    float4 a = *ap;
    a.x += hv.x * w; a.y += hv.y * w; a.z += hv.z * w; a.w += hv.w * w;
    *ap = a;
  }
}

// out = (add ? out : 0) + 0.5 * relu(acc + bias)
__global__ void k_finalize(const float* __restrict__ acc,
                           const float* __restrict__ bias,
                           float* __restrict__ out, int n, int add) {
  int t = blockIdx.x * blockDim.x + threadIdx.x;
  int i = t >> 4;
  int q = t & 15;
  if (i < n) {
    const float4 a = *(const float4*)(acc + (size_t)i * D_FEAT + q * 4);
    const float4 b = *(const float4*)(bias + q * 4);
    float4 r;
    r.x = fmaxf(a.x + b.x, 0.f) * 0.5f;
    r.y = fmaxf(a.y + b.y, 0.f) * 0.5f;
    r.z = fmaxf(a.z + b.z, 0.f) * 0.5f;
    r.w = fmaxf(a.w + b.w, 0.f) * 0.5f;
    float4* op = (float4*)(out + (size_t)i * D_FEAT + q * 4);
    if (add) {
      float4 o = *op;
      o.x += r.x; o.y += r.y; o.z += r.z; o.w += r.w;
      *op = o;
    } else {
      *op = r;
    }
  }
}

extern "C" void kernel_launch(void* const* d_in, const int* in_sizes, int n_in,
                              void* d_out, int out_size, void* d_ws, size_t ws_size,
                              hipStream_t stream) {
  const float*     x    = (const float*)d_in[0];
  const long long* edge = (const long long*)d_in[1];  // int64 [2, E] flat
  const float*     W_f  = (const float*)d_in[2];
  const float*     b_f  = (const float*)d_in[3];
  const float*     W_b  = (const float*)d_in[4];
  const float*     b_b  = (const float*)d_in[5];
  float* out = (float*)d_out;

  const int N = in_sizes[0] / D_FEAT;   // 100000
  const int E = in_sizes[1] / 2;        // 1000000

  // Workspace layout (floats): dinv_f[N] | dinv_b[N] | H[N*64] | acc[N*64]
  float* ws     = (float*)d_ws;
  float* dinv_f = ws;
  float* dinv_b = ws + N;
  float* H      = ws + 2 * (size_t)N;
  float* acc    = H + (size_t)N * D_FEAT;

  const long long* esrc = edge;
  const long long* edst = edge + E;

  const int B = 256;
  const int gDeg  = (2 * N + B - 1) / B;
  const int gEdg  = (E + B - 1) / B;
  const int gVec  = (N * 16 + B - 1) / B;                       // float4 over N*64
  const int gScat = (int)(((long long)E * 16 + B - 1) / B);
  const int gGemm = N / 16;                                     // 100000 % 16 == 0

  // ---- degrees: deg = 1 (self-loop) + histogram; then dinv = rsqrt(deg) ----
  k_fill1<<<gDeg, B, 0, stream>>>(dinv_f, 2 * N);               // fills both arrays
  k_count_deg<<<gEdg, B, 0, stream>>>(esrc, edst, dinv_f, dinv_b, E);
  k_rsqrt<<<gDeg, B, 0, stream>>>(dinv_f, 2 * N);

  // ---- direction 1: messages src -> dst, norm from in-degree ----
  k_zero4<<<gVec, B, 0, stream>>>((float4*)acc, N * 16);
  k_gemm_wmma_f32<<<gGemm, 128, 0, stream>>>(x, W_f, H);
  k_scatter<<<gScat, B, 0, stream>>>(esrc, edst, dinv_f, H, acc, E);
  k_selfloop<<<gVec, B, 0, stream>>>(dinv_f, H, acc, N);
  k_finalize<<<gVec, B, 0, stream>>>(acc, b_f, out, N, /*add=*/0);

  // ---- direction 2: messages dst -> src, norm from out-degree ----
  k_zero4<<<gVec, B, 0, stream>>>((float4*)acc, N * 16);
  k_gemm_wmma_f32<<<gGemm, 128, 0, stream>>>(x, W_b, H);
  k_scatter<<<gScat, B, 0, stream>>>(edst, esrc, dinv_b, H, acc, E);
  k_selfloop<<<gVec, B, 0, stream>>>(dinv_b, H, acc, N);
  k_finalize<<<gVec, B, 0, stream>>>(acc, b_b, out, N, /*add=*/1);
}